// SIRNetBase_2757369004524
// MI455X (gfx1250) — compile-verified
//
#include <hip/hip_runtime.h>

// Problem constants (from the reference)
#define T_STEPS 2048
#define B_SIZE  8192
#define FEAT    6

typedef __attribute__((ext_vector_type(4))) float v4f;

#if defined(__HIP_DEVICE_COMPILE__)
typedef __attribute__((address_space(1))) int as1_int;  // global
typedef __attribute__((address_space(3))) int as3_int;  // LDS
#endif

// ---------------------------------------------------------------------------
// Kernel 1: per-timestep sum-of-squares reduction -> b_seq[t] = q * ||X[t]||
// One block per timestep. Row = B*6 = 49152 floats = 12288 float4.
// Streaming read of 403 MB; non-temporal b128 loads + prefetch so we do not
// pollute L2 (data is read exactly once).
// ---------------------------------------------------------------------------
__global__ __launch_bounds__(256) void sir_rownorm_kernel(
    const float* __restrict__ X, const float* __restrict__ q,
    float* __restrict__ b_seq) {
  const int t   = blockIdx.x;
  const int tid = threadIdx.x;
  const v4f* row = (const v4f*)(X + (size_t)t * (size_t)(B_SIZE * FEAT));
  const int NV = (B_SIZE * FEAT) / 4;  // 12288 float4 per row

  float acc = 0.0f;
  for (int i = tid; i < NV; i += 256) {
    // gfx1250: global_prefetch_b8 (speculative, address validity irrelevant)
    __builtin_prefetch(&row[i + 2 * 256], 0, 0);
    v4f v = __builtin_nontemporal_load(&row[i]);  // global_load_b128 th:NT
    acc = fmaf(v.x, v.x, acc);
    acc = fmaf(v.y, v.y, acc);
    acc = fmaf(v.z, v.z, acc);
    acc = fmaf(v.w, v.w, acc);
  }

  __shared__ float red[256];
  red[tid] = acc;
  __syncthreads();
#pragma unroll
  for (int s = 128; s > 0; s >>= 1) {
    if (tid < s) red[tid] += red[tid + s];
    __syncthreads();
  }
  if (tid == 0) b_seq[t] = q[0] * sqrtf(red[0]);
}

// ---------------------------------------------------------------------------
// Kernel 2: SIR scan. One lane per batch element; 256 blocks of one wave32
// each to spread the latency-bound dependency chain across all SIMDs.
// b_seq (8 KB) is staged into LDS via the gfx1250 async-to-LDS path
// (ASYNCcnt + s_wait_asynccnt). Outputs are streamed with non-temporal
// stores (268 MB write-once stream).
// ---------------------------------------------------------------------------
__global__ __launch_bounds__(32) void sir_scan_kernel(
    const float* __restrict__ b_seq, const float* __restrict__ kparam,
    const float* __restrict__ i0, float* __restrict__ hiddens,
    float* __restrict__ outputs) {
  __shared__ float lds_b[T_STEPS];  // 8 KB
  const int tid = threadIdx.x;

#if defined(__HIP_DEVICE_COMPILE__) && defined(__gfx1250__) && \
    __has_builtin(__builtin_amdgcn_global_load_async_to_lds_b32)
  // gfx1250 async DMA: global -> LDS without touching VGPR data.
  for (int i = tid; i < T_STEPS; i += 32) {
    __builtin_amdgcn_global_load_async_to_lds_b32(
        (as1_int*)(b_seq + i), (as3_int*)(lds_b + i),
        /*imm_offset=*/0, /*cpol=*/0);
  }
#if __has_builtin(__builtin_amdgcn_s_wait_asynccnt)
  __builtin_amdgcn_s_wait_asynccnt(0);
#else
  asm volatile("s_wait_asynccnt 0" ::: "memory");
#endif
#else
  for (int i = tid; i < T_STEPS; i += 32) lds_b[i] = b_seq[i];
#endif
  __syncthreads();  // single-wave block: cheap

  const int b = blockIdx.x * 32 + tid;  // 0..8191
  const float kk  = kparam[0];
  const float i0s = i0[0];

  float I = i0s;
  float R = 0.0f;
  float S = 1.0f - i0s;

  const size_t hstep = (size_t)B_SIZE * 3;
  float* __restrict__ hptr = hiddens + (size_t)b * 3;
  float* __restrict__ optr = outputs + b;

  for (int t = 0; t < T_STEPS; ++t) {
    const float bb  = lds_b[t];       // uniform ds_load, broadcast
    const float si  = S * I;
    const float rec = kk * I;
    const float nI  = fmaf(bb, si, I) - rec;   // I + b*S*I - k*I
    const float nR  = R + rec;                 // R + k*I
    const float nS  = fmaf(-bb, si, S);        // S - b*S*I
    const float out = nI + nR;

    // Lane b writes 3 contiguous floats at byte offset 12*b within the row:
    // the wave's 32 lanes cover a contiguous 384 B span -> fully coalesced.
    __builtin_nontemporal_store(nI, hptr + 0);
    __builtin_nontemporal_store(nR, hptr + 1);
    __builtin_nontemporal_store(nS, hptr + 2);
    __builtin_nontemporal_store(out, optr);

    I = nI; R = nR; S = nS;
    hptr += hstep;
    optr += B_SIZE;
  }
}

// ---------------------------------------------------------------------------
// Host-side launcher. Inputs: X (T*B*6 f32), q (1), k (1), i0 (1).
// d_out = hiddens (T*B*3) ++ outputs (T*B). d_ws holds b_seq (T floats).
// ---------------------------------------------------------------------------
extern "C" void kernel_launch(void* const* d_in, const int* in_sizes, int n_in,
                              void* d_out, int out_size, void* d_ws,
                              size_t ws_size, hipStream_t stream) {
  (void)in_sizes; (void)n_in; (void)out_size; (void)ws_size;
  const float* X  = (const float*)d_in[0];
  const float* q  = (const float*)d_in[1];
  const float* k  = (const float*)d_in[2];
  const float* i0 = (const float*)d_in[3];

  float* hiddens = (float*)d_out;
  float* outputs = (float*)d_out + (size_t)T_STEPS * (size_t)B_SIZE * 3;
  float* b_seq   = (float*)d_ws;  // T_STEPS floats

  sir_rownorm_kernel<<<T_STEPS, 256, 0, stream>>>(X, q, b_seq);
  sir_scan_kernel<<<B_SIZE / 32, 32, 0, stream>>>(b_seq, k, i0, hiddens,
                                                  outputs);
}